// TripletLoss_75943611728236
// MI455X (gfx1250) — compile-verified
//
#include <hip/hip_runtime.h>
#include <hip/hip_bf16.h>
#include <math.h>

typedef __attribute__((ext_vector_type(2))) float v2f;
typedef __attribute__((ext_vector_type(8))) float v8f;
typedef __attribute__((ext_vector_type(4))) int   v4i;

#define BS     2048
#define DIM    256
#define TILE   128
#define KCH    16          // K-chunk staged in LDS (double-buffered)
#define LST    20          // LDS row stride in floats: 80B rows -> 16B aligned, conflict-free
#define NCHUNK (DIM / KCH)
#define MARGIN 0.3f

#if __has_builtin(__builtin_amdgcn_global_load_async_to_lds_b128)
#define HAVE_ASYNC 1
typedef __attribute__((address_space(1))) v4i glob_v4i;   // global (AS1) int4
typedef __attribute__((address_space(3))) v4i lds_v4i;    // LDS    (AS3) int4
#else
#define HAVE_ASYNC 0
#endif

__device__ __forceinline__ void wait_async_copies() {
#if HAVE_ASYNC
#if __has_builtin(__builtin_amdgcn_s_wait_asynccnt)
    __builtin_amdgcn_s_wait_asynccnt(0);
#else
    asm volatile("s_wait_asynccnt 0x0" ::: "memory");
#endif
#endif
}

// ---------------------------------------------------------------------------
// Kernel 1: per-row norms. One block per row, 256 threads = 256 dims.
// ---------------------------------------------------------------------------
__global__ __launch_bounds__(256) void norms_kernel(const float* __restrict__ sketch,
                                                    const float* __restrict__ photo,
                                                    float* __restrict__ pos,
                                                    float* __restrict__ sn,
                                                    float* __restrict__ pn) {
    __shared__ float r0[256], r1[256], r2[256];
    const int row = blockIdx.x;
    const int t   = threadIdx.x;
    const float s = sketch[(size_t)row * DIM + t];
    const float p = photo[(size_t)row * DIM + t];
    const float d = s - p;
    r0[t] = s * s;
    r1[t] = p * p;
    r2[t] = d * d;
    __syncthreads();
    for (int st = 128; st > 0; st >>= 1) {
        if (t < st) { r0[t] += r0[t + st]; r1[t] += r1[t + st]; r2[t] += r2[t + st]; }
        __syncthreads();
    }
    if (t == 0) {
        sn[row]  = r0[0];
        pn[row]  = r1[0];
        pos[row] = sqrtf(r2[0]);
    }
}

// ---------------------------------------------------------------------------
// Kernel 2: fused 128x128 tile of cross = photo * sketch^T via f32 WMMA.
// 8 wave32 arranged 4x2: each wave owns a 32x64 patch (2 A frags x 4 B frags
// -> 8 accumulators). Global->LDS staging uses GLOBAL_LOAD_ASYNC_TO_LDS_B128
// with LDS double buffering (ASYNCcnt) when available.
// ---------------------------------------------------------------------------
__global__ __launch_bounds__(256) void tile_kernel(const float* __restrict__ sketch,
                                                   const float* __restrict__ photo,
                                                   const float* __restrict__ pos,
                                                   const float* __restrict__ sn,
                                                   const float* __restrict__ pn,
                                                   float* __restrict__ partials) {
    // [buf][arr][row*LST + col]: arr 0 = photo (A/i), arr 1 = sketch (B/j)
    __shared__ float ldsbuf[2][2][TILE * LST];
    __shared__ float red[256];

    const int tid  = threadIdx.x;
    const int wave = tid >> 5;        // 0..7
    const int lane = tid & 31;
    const int l16  = lane & 15;
    const int lhi  = lane >> 4;       // 0 or 1
    const int wr   = wave >> 1;       // row group 0..3  (32 rows each)
    const int wc   = wave & 1;        // col group 0..1  (64 cols each)

    const int bi = blockIdx.x & 15;   // column tile (sketch rows j)
    const int bj = blockIdx.x >> 4;   // row tile    (photo rows i)
    const int rowBase = bj * TILE;
    const int colBase = bi * TILE;

    v8f acc[2][4];
#pragma unroll
    for (int ar = 0; ar < 2; ++ar)
#pragma unroll
        for (int bc = 0; bc < 4; ++bc)
            acc[ar][bc] = (v8f){0.f, 0.f, 0.f, 0.f, 0.f, 0.f, 0.f, 0.f};

    // ---- stage one 128 x KCH chunk of each operand into LDS buffer `buf` ----
    auto issue_copy = [&](int kc, int buf) {
#pragma unroll
        for (int it = 0; it < 2; ++it) {
            const int linear = tid + it * 256;      // float4 index 0..511
            const int r  = linear >> 2;             // row 0..127
            const int c4 = (linear & 3) << 2;       // col 0,4,8,12
            const float* gp = photo  + (size_t)(rowBase + r) * DIM + kc + c4;
            const float* gs = sketch + (size_t)(colBase + r) * DIM + kc + c4;
            float* lp = &ldsbuf[buf][0][r * LST + c4];
            float* ls = &ldsbuf[buf][1][r * LST + c4];
#if HAVE_ASYNC
            __builtin_amdgcn_global_load_async_to_lds_b128((glob_v4i*)gp, (lds_v4i*)lp, 0, 0);
            __builtin_amdgcn_global_load_async_to_lds_b128((glob_v4i*)gs, (lds_v4i*)ls, 0, 0);
#else
            *(float4*)lp = *(const float4*)gp;
            *(float4*)ls = *(const float4*)gs;
#endif
        }
    };

    issue_copy(0, 0);
    wait_async_copies();
    __syncthreads();

    for (int ch = 0; ch < NCHUNK; ++ch) {
        const int buf = ch & 1;
        if (ch + 1 < NCHUNK) issue_copy((ch + 1) * KCH, buf ^ 1);

        const float* lp = ldsbuf[buf][0];
        const float* ls = ldsbuf[buf][1];
#pragma unroll
        for (int kk = 0; kk < KCH; kk += 4) {
            const int kcol = kk + (lhi << 1);       // lane holds K = 2*lhi + {0,1}
            // group all fragment loads so the compiler can clause + wait once
            const v2f a0 = *(const v2f*)(&lp[(wr * 32 +      l16) * LST + kcol]);
            const v2f a1 = *(const v2f*)(&lp[(wr * 32 + 16 + l16) * LST + kcol]);
            const v2f b0 = *(const v2f*)(&ls[(wc * 64 +      l16) * LST + kcol]);
            const v2f b1 = *(const v2f*)(&ls[(wc * 64 + 16 + l16) * LST + kcol]);
            const v2f b2 = *(const v2f*)(&ls[(wc * 64 + 32 + l16) * LST + kcol]);
            const v2f b3 = *(const v2f*)(&ls[(wc * 64 + 48 + l16) * LST + kcol]);
            acc[0][0] = __builtin_amdgcn_wmma_f32_16x16x4_f32(false, a0, false, b0, (short)0, acc[0][0], false, false);
            acc[0][1] = __builtin_amdgcn_wmma_f32_16x16x4_f32(false, a0, false, b1, (short)0, acc[0][1], false, false);
            acc[0][2] = __builtin_amdgcn_wmma_f32_16x16x4_f32(false, a0, false, b2, (short)0, acc[0][2], false, false);
            acc[0][3] = __builtin_amdgcn_wmma_f32_16x16x4_f32(false, a0, false, b3, (short)0, acc[0][3], false, false);
            acc[1][0] = __builtin_amdgcn_wmma_f32_16x16x4_f32(false, a1, false, b0, (short)0, acc[1][0], false, false);
            acc[1][1] = __builtin_amdgcn_wmma_f32_16x16x4_f32(false, a1, false, b1, (short)0, acc[1][1], false, false);
            acc[1][2] = __builtin_amdgcn_wmma_f32_16x16x4_f32(false, a1, false, b2, (short)0, acc[1][2], false, false);
            acc[1][3] = __builtin_amdgcn_wmma_f32_16x16x4_f32(false, a1, false, b3, (short)0, acc[1][3], false, false);
        }

        wait_async_copies();   // next chunk landed in the other buffer
        __syncthreads();
    }

    // ---- epilogue: tl = relu(pos[j] - sqrt(max(pn[i]+sn[j]-2*cross,0)) + m) ----
    float lsum = 0.f;
#pragma unroll
    for (int ar = 0; ar < 2; ++ar) {
#pragma unroll
        for (int bc = 0; bc < 4; ++bc) {
            const int j = colBase + wc * 64 + bc * 16 + l16;     // C/D: N = lane%16
            const float posj = pos[j];
            const float snj  = sn[j];
#pragma unroll
            for (int v = 0; v < 8; ++v) {
                const int i = rowBase + wr * 32 + ar * 16 + (lhi << 3) + v;  // M = v + 8*(lane/16)
                const float cr = acc[ar][bc][v];
                float n2 = pn[i] + snj - 2.f * cr;
                n2 = fmaxf(n2, 0.f);
                float t = posj - sqrtf(n2) + MARGIN;
                t = (i == j) ? 0.f : fmaxf(t, 0.f);
                lsum += t;
            }
        }
    }

    // deterministic block reduction
    red[tid] = lsum;
    __syncthreads();
    for (int st = 128; st > 0; st >>= 1) {
        if (tid < st) red[tid] += red[tid + st];
        __syncthreads();
    }
    if (tid == 0) partials[blockIdx.x] = red[0];
}

// ---------------------------------------------------------------------------
// Kernel 3: reduce 256 partials -> mean
// ---------------------------------------------------------------------------
__global__ __launch_bounds__(256) void reduce_kernel(const float* __restrict__ partials,
                                                     float* __restrict__ out) {
    __shared__ float red[256];
    const int tid = threadIdx.x;
    red[tid] = partials[tid];
    __syncthreads();
    for (int st = 128; st > 0; st >>= 1) {
        if (tid < st) red[tid] += red[tid + st];
        __syncthreads();
    }
    if (tid == 0) out[0] = red[0] / ((float)BS * (float)BS);
}

extern "C" void kernel_launch(void* const* d_in, const int* in_sizes, int n_in,
                              void* d_out, int out_size, void* d_ws, size_t ws_size,
                              hipStream_t stream) {
    (void)in_sizes; (void)n_in; (void)out_size; (void)ws_size;
    const float* sketch = (const float*)d_in[0];
    const float* photo  = (const float*)d_in[1];
    float* out = (float*)d_out;

    float* w        = (float*)d_ws;
    float* pos      = w;            // 2048
    float* sn       = w + BS;       // 2048
    float* pn       = w + 2 * BS;   // 2048
    float* partials = w + 3 * BS;   // 256

    norms_kernel<<<BS, 256, 0, stream>>>(sketch, photo, pos, sn, pn);
    tile_kernel<<<(BS / TILE) * (BS / TILE), 256, 0, stream>>>(sketch, photo, pos, sn, pn, partials);
    reduce_kernel<<<1, 256, 0, stream>>>(partials, out);
}